// GPTMultiheadMaskedAttention_39608188404432
// MI455X (gfx1250) — compile-verified
//
#include <hip/hip_runtime.h>
#include <hip/hip_bf16.h>

// ---------------------------------------------------------------------------
// GPT multi-head causal attention, fp32, built on CDNA5 V_WMMA_F32_16X16X4_F32
// with GLOBAL_LOAD_ASYNC_TO_LDS_B128 tile staging (ASYNCcnt path).
//   S = 4096, HIDDEN = 1024, NUM_HEADS = 16, HEAD_DIM = 64
// ---------------------------------------------------------------------------

typedef __attribute__((ext_vector_type(2))) float v2f;
typedef __attribute__((ext_vector_type(8))) float v8f;

// D = A(16x4) * B(4x16) + C(16x16), all fp32, one matrix per wave32.
#define WMMA_F32x4(a, b, c)                                                   \
  __builtin_amdgcn_wmma_f32_16x16x4_f32(false, (a), false, (b), (short)0,     \
                                        (c), false, false)

__device__ __forceinline__ void lds_fence() {
  asm volatile("s_wait_dscnt 0" ::: "memory");
}

__device__ __forceinline__ void async_fence() {
  asm volatile("s_wait_asynccnt 0" ::: "memory");
}

// Async DMA of 16 bytes global -> LDS, no VGPR staging, tracked by ASYNCcnt.
// Generic pointers to __shared__ are {aperture_hi, lds_offset}; the low 32
// bits are exactly the DS byte address the async unit consumes.
__device__ __forceinline__ void async_copy_b128(void* lds_dst,
                                                const void* gsrc) {
  const unsigned int lds_off = (unsigned int)(unsigned long long)lds_dst;
  const unsigned long long ga = (unsigned long long)gsrc;
  asm volatile("global_load_async_to_lds_b128 %0, %1, off"
               :
               : "v"(lds_off), "v"(ga)
               : "memory");
}

// ---------------------------------------------------------------------------
// Generic C[M,N] = A[M,K] * B[K,N] + bias[N]  (row-major, M%128==N%64==K%32==0)
// Block: 128 threads = 4 waves; block tile 128x64; wave tile 32x64.
// K staged through LDS in 32-wide chunks via async-to-LDS b128 DMA.
// ---------------------------------------------------------------------------
__global__ __launch_bounds__(128)
void gemm_bias_wmma(const float* __restrict__ A, const float* __restrict__ B,
                    const float* __restrict__ bias, float* __restrict__ C,
                    int M, int N, int K) {
  __shared__ float As[128 * 32];  // 16 KB
  __shared__ float Bs[32 * 64];   //  8 KB

  const int tid  = threadIdx.x;
  const int wave = tid >> 5;
  const int lane = tid & 31;
  const int lrow = lane & 15;   // M index (A/D) or N index (B/D)
  const int half = lane >> 4;   // 0: K=0,1   1: K=2,3
  const int koff = half * 2;
  const int row0 = blockIdx.y * 128;
  const int col0 = blockIdx.x * 64;

  v8f acc[2][4];
#pragma unroll
  for (int mt = 0; mt < 2; ++mt)
#pragma unroll
    for (int nt = 0; nt < 4; ++nt)
#pragma unroll
      for (int i = 0; i < 8; ++i) acc[mt][nt][i] = 0.0f;

  for (int k0 = 0; k0 < K; k0 += 32) {
    // Stage A tile 128x32 (4096 floats): 8 async b128 DMAs per thread
#pragma unroll
    for (int it = 0; it < 8; ++it) {
      const int idx = (tid + it * 128) * 4;  // float index 0..4095
      const int r = idx >> 5, c = idx & 31;
      async_copy_b128(&As[r * 32 + c], &A[(size_t)(row0 + r) * K + k0 + c]);
    }
    // Stage B tile 32x64 (2048 floats): 4 async b128 DMAs per thread
#pragma unroll
    for (int it = 0; it < 4; ++it) {
      const int idx = (tid + it * 128) * 4;
      const int r = idx >> 6, c = idx & 63;
      async_copy_b128(&Bs[r * 64 + c], &B[(size_t)(k0 + r) * N + col0 + c]);
    }
    async_fence();    // this wave's DMAs landed in LDS
    __syncthreads();  // everyone's DMAs visible to everyone

#pragma unroll
    for (int kk = 0; kk < 32; kk += 4) {
      v2f a0, a1;  // two A fragments (rows wave*32+[0,16)+lrow)
      a0.x = As[(wave * 32 + lrow) * 32 + kk + koff];
      a0.y = As[(wave * 32 + lrow) * 32 + kk + koff + 1];
      a1.x = As[(wave * 32 + 16 + lrow) * 32 + kk + koff];
      a1.y = As[(wave * 32 + 16 + lrow) * 32 + kk + koff + 1];
#pragma unroll
      for (int nt = 0; nt < 4; ++nt) {
        v2f b;  // B fragment: 4x16, shared by both A fragments
        b.x = Bs[(kk + koff) * 64 + nt * 16 + lrow];
        b.y = Bs[(kk + koff + 1) * 64 + nt * 16 + lrow];
        acc[0][nt] = WMMA_F32x4(a0, b, acc[0][nt]);
        acc[1][nt] = WMMA_F32x4(a1, b, acc[1][nt]);
      }
    }
    __syncthreads();  // all reads done before next chunk's DMA overwrites
  }

  // Epilogue: C/D layout -> VGPR i holds row (i + half*8), col = lrow
#pragma unroll
  for (int mt = 0; mt < 2; ++mt) {
#pragma unroll
    for (int nt = 0; nt < 4; ++nt) {
      const int cg = col0 + nt * 16 + lrow;
      const float bv = bias[cg];
#pragma unroll
      for (int i = 0; i < 8; ++i) {
        const int rg = row0 + wave * 32 + mt * 16 + i + half * 8;
        C[(size_t)rg * N + cg] = acc[mt][nt][i] + bv;
      }
    }
  }
}

// ---------------------------------------------------------------------------
// Flash attention, fp32 WMMA. One wave handles one (head, 16-query block).
// QKV row layout: [S][3*H]  with Q at col 0, K at col H, V at col 2H.
// AV output: [S][H].
// ---------------------------------------------------------------------------
__global__ __launch_bounds__(128)
void attention_wmma(const float* __restrict__ QKV, float* __restrict__ AV,
                    int S, int ld /*3H*/, int H) {
  __shared__ float Pb[4][16 * 16];  // per-wave P staging (C-layout -> A-layout)

  const int tid  = threadIdx.x;
  const int wave = tid >> 5;
  const int lane = tid & 31;
  const int lrow = lane & 15;
  const int half = lane >> 4;
  const int koff = half * 2;

  const int h    = blockIdx.y;
  const int qb   = blockIdx.x * 4 + wave;  // query block (wave-uniform)
  const int q0   = qb * 16;
  const int qcol = h * 64;
  const int kcol = H + h * 64;
  const int vcol = 2 * H + h * 64;
  float* P = Pb[wave];

  // Q fragments held in registers for the whole pass: 16 x v2f = 32 VGPRs
  v2f qa[16];
#pragma unroll
  for (int kk8 = 0; kk8 < 16; ++kk8) {
    const int kk = kk8 * 4 + koff;
    const float* qp = &QKV[(size_t)(q0 + lrow) * ld + qcol + kk];
    qa[kk8].x = qp[0];
    qa[kk8].y = qp[1];
  }

  float m[8], lsum[8];
  v8f o[4];
#pragma unroll
  for (int i = 0; i < 8; ++i) { m[i] = -3.0e38f; lsum[i] = 0.0f; }
#pragma unroll
  for (int nt = 0; nt < 4; ++nt)
#pragma unroll
    for (int i = 0; i < 8; ++i) o[nt][i] = 0.0f;

  const float scale = 0.125f;  // 1/sqrt(64)

  for (int kb = 0; kb <= qb; ++kb) {  // causal: only lower-triangular blocks
    const int key0 = kb * 16;

    // S(16x16) = Q(16x64) * K^T(64x16); B[k][n] = K[key0+n][k]
    v8f s = {};
#pragma unroll
    for (int kk8 = 0; kk8 < 16; ++kk8) {
      const int kk = kk8 * 4 + koff;
      const float* kp = &QKV[(size_t)(key0 + lrow) * ld + kcol + kk];
      v2f b;
      b.x = kp[0];
      b.y = kp[1];
      s = WMMA_F32x4(qa[kk8], b, s);
    }

    // Online softmax. Row of element s[i]: q0+i+half*8, col: key0+lrow.
    float p[8];
#pragma unroll
    for (int i = 0; i < 8; ++i) {
      const int rg = q0 + i + half * 8;
      const int cg = key0 + lrow;
      float sv = s[i] * scale;
      if (cg > rg) sv = -1.0e6f;  // causal mask, matches reference NEG_INF

      float rmax = sv;  // reduce across the 16-lane N group (halves disjoint)
      rmax = fmaxf(rmax, __shfl_xor(rmax, 1));
      rmax = fmaxf(rmax, __shfl_xor(rmax, 2));
      rmax = fmaxf(rmax, __shfl_xor(rmax, 4));
      rmax = fmaxf(rmax, __shfl_xor(rmax, 8));
      const float mn = fmaxf(m[i], rmax);
      const float pe = __expf(sv - mn);
      float rsum = pe;
      rsum += __shfl_xor(rsum, 1);
      rsum += __shfl_xor(rsum, 2);
      rsum += __shfl_xor(rsum, 4);
      rsum += __shfl_xor(rsum, 8);
      const float alpha = __expf(m[i] - mn);  // first iter: exp(-inf) = 0
      lsum[i] = lsum[i] * alpha + rsum;
      m[i] = mn;
      p[i] = pe;
#pragma unroll
      for (int nt = 0; nt < 4; ++nt) o[nt][i] *= alpha;
    }

    // Stage P (C-layout) into LDS, read back in A-fragment layout.
#pragma unroll
    for (int i = 0; i < 8; ++i) P[(i + half * 8) * 16 + lrow] = p[i];
    lds_fence();

    // O(16x64) += P(16x16) * V(16x64); B[k][n] = V[key0+k][vcol+n]
#pragma unroll
    for (int kk4 = 0; kk4 < 4; ++kk4) {
      const int kk = kk4 * 4 + koff;
      v2f pa;
      pa.x = P[lrow * 16 + kk];
      pa.y = P[lrow * 16 + kk + 1];
      const float* v0 = &QKV[(size_t)(key0 + kk) * ld + vcol];
      const float* v1 = &QKV[(size_t)(key0 + kk + 1) * ld + vcol];
#pragma unroll
      for (int nt = 0; nt < 4; ++nt) {
        v2f vb;
        vb.x = v0[nt * 16 + lrow];
        vb.y = v1[nt * 16 + lrow];
        o[nt] = WMMA_F32x4(pa, vb, o[nt]);
      }
    }
    lds_fence();  // P reads done before next block overwrites it
  }

  // Normalize and write AV[S][H]
#pragma unroll
  for (int i = 0; i < 8; ++i) {
    const float inv = 1.0f / lsum[i];
    const int rg = q0 + i + half * 8;
#pragma unroll
    for (int nt = 0; nt < 4; ++nt) {
      AV[(size_t)rg * H + h * 64 + nt * 16 + lrow] = o[nt][i] * inv;
    }
  }
}

// ---------------------------------------------------------------------------
extern "C" void kernel_launch(void* const* d_in, const int* in_sizes, int n_in,
                              void* d_out, int out_size, void* d_ws,
                              size_t ws_size, hipStream_t stream) {
  const float* x     = (const float*)d_in[0];  // [4096,1024]
  const float* W_qkv = (const float*)d_in[1];  // [1024,3072]
  const float* b_qkv = (const float*)d_in[2];  // [3072]
  const float* W_o   = (const float*)d_in[3];  // [1024,1024]
  const float* b_o   = (const float*)d_in[4];  // [1024]
  float* out = (float*)d_out;                  // [4096,1024]

  const int S = 4096, H = 1024, NH = 16;

  float* qkv = (float*)d_ws;                   // [4096,3072] = 48 MiB
  float* av  = qkv + (size_t)S * 3 * H;        // [4096,1024] = 16 MiB

  // 1) QKV projection: [S,H] @ [H,3H] + b
  gemm_bias_wmma<<<dim3(3 * H / 64, S / 128), 128, 0, stream>>>(
      x, W_qkv, b_qkv, qkv, S, 3 * H, H);

  // 2) Causal flash attention per head
  attention_wmma<<<dim3(S / 16 / 4, NH), 128, 0, stream>>>(qkv, av, S, 3 * H, H);

  // 3) Output projection: [S,H] @ [H,H] + b
  gemm_bias_wmma<<<dim3(H / 64, S / 128), 128, 0, stream>>>(
      av, W_o, b_o, out, S, H, H);
}